// Modified_Hausdorff_Distance_35416300323047
// MI455X (gfx1250) — compile-verified
//
#include <hip/hip_runtime.h>
#include <math.h>

#define N_    2
#define C_    4
#define H_    112
#define W_    112
#define HW_   (H_ * W_)
#define QT_   784            // HW_/16 = max query tiles per (pair, dir)
#define NPAIR 6              // N_ * (C_-1)
#define SENT  1.0e4f         // sentinel coordinate for list padding
#define FINF  3.0e38f

typedef __attribute__((ext_vector_type(2))) float v2f;
typedef __attribute__((ext_vector_type(8))) float v8f;

// ---------------------------------------------------------------------------
// ws layout (bytes):
//   [0,      48)     int counts[12]         A counts p=0..5, B counts 6..11
//   [256,    37888) float partials[12*QT_] per-block partial sums (fixed slots)
//   [37888, ...)    float2 lists[12*HW_]   A lists at p*HW_, B lists at (6+p)*HW_
// ---------------------------------------------------------------------------

__global__ void hd_init_kernel(float2* __restrict__ lists, int* __restrict__ counts) {
    int idx = blockIdx.x * blockDim.x + threadIdx.x;
    if (idx < 12 * HW_) lists[idx] = make_float2(SENT, SENT);
    if (idx < 12) counts[idx] = 0;
}

__global__ void hd_compact_kernel(const float* __restrict__ pred,
                                  const int* __restrict__ labels,
                                  float2* __restrict__ lists,
                                  int* __restrict__ counts) {
    int idx = blockIdx.x * blockDim.x + threadIdx.x;
    if (idx >= N_ * HW_) return;
    int b   = idx / HW_;
    int pix = idx - b * HW_;
    int h   = pix / W_;
    int w   = pix - h * W_;
    float2 pt = make_float2((float)h, (float)w);

    // softmax is monotone per-pixel => argmax of logits; strict '>' = first max
    int base   = (b * C_) * HW_ + pix;
    float best = pred[base];
    int   bc   = 0;
#pragma unroll
    for (int ch = 1; ch < C_; ++ch) {
        float v = pred[base + ch * HW_];
        if (v > best) { best = v; bc = ch; }
    }
    if (bc >= 1) {
        int p   = b * (C_ - 1) + (bc - 1);
        int pos = atomicAdd(&counts[p], 1);
        lists[p * HW_ + pos] = pt;
    }
#pragma unroll
    for (int j = 1; j < C_; ++j) {
        if (labels[base + j * HW_] == 1) {
            int p   = b * (C_ - 1) + (j - 1);
            int pos = atomicAdd(&counts[NPAIR + p], 1);
            lists[(NPAIR + p) * HW_ + pos] = pt;
        }
    }
}

// One wave32 block per (query tile, pair, direction).
// d^2(q,p) = |q|^2 + |p|^2 - 2 q.p computed as a 16x16 tile via one
// V_WMMA_F32_16X16X4_F32, then v_min per tile into running minima.
__global__ __launch_bounds__(32)
void hd_min_kernel(const float2* __restrict__ lists,
                   const int* __restrict__ counts,
                   float* __restrict__ partials) {
    const int tile = blockIdx.x;
    const int y    = blockIdx.y;     // y = pair*2 + dir
    const int p    = y >> 1;
    const int dir  = y & 1;
    const int lane = threadIdx.x;
    const int nl   = lane & 15;
    const int hi   = lane >> 4;

    const int cq = (dir == 0) ? counts[p] : counts[NPAIR + p];
    const int cs = (dir == 0) ? counts[NPAIR + p] : counts[p];
    const float2* Q = lists + (size_t)((dir == 0) ? p : (NPAIR + p)) * HW_;
    const float2* S = lists + (size_t)((dir == 0) ? (NPAIR + p) : p) * HW_;

    if (tile * 16 >= cq) {                 // unused tile: must still zero its slot
        if (lane == 0) partials[y * QT_ + tile] = 0.0f;
        return;
    }

    __shared__ float s_qn[16];
    __shared__ float s_min[256];

    // Load query tile (16 points, both wave halves read the same 16).
    float2 q  = Q[tile * 16 + nl];
    float  qn = q.x * q.x + q.y * q.y;
    // A-matrix 16x4 (documented layout): lanes 0-15 -> K=0,1 ; lanes 16-31 -> K=2,3.
    // Zero the K=2,3 half so only K=0,1 carry (-2r, -2c).
    const float a0 = hi ? 0.0f : (-2.0f * q.x);
    const float a1 = hi ? 0.0f : (-2.0f * q.y);
    if (lane < 16) s_qn[nl] = qn;
    __syncthreads();

    // C/D layout: VGPR v, lane -> element (m = v + 8*hi, n = lane&15)
    float qnv[8];
#pragma unroll
    for (int v = 0; v < 8; ++v) qnv[v] = s_qn[v + 8 * hi];

    v8f minv;
#pragma unroll
    for (int v = 0; v < 8; ++v) minv[v] = FINF;

    const v2f A = { a0, a1 };
    const int stiles = (cs + 15) >> 4;
    for (int t = 0; t < stiles; ++t) {
        float2 s  = S[t * 16 + nl];
        float  pn = s.x * s.x + s.y * s.y;
        // B-matrix 4x16: fill BOTH lane halves with (r_p, c_p). Whichever half
        // maps to K=0,1 multiplies A's documented K=0,1 operands; the half
        // mapped to K=2,3 multiplies A's zeros. Correct under either B layout.
        v2f B = { s.x, s.y };
        v8f Cm;
#pragma unroll
        for (int v = 0; v < 8; ++v) Cm[v] = qnv[v] + pn;
        // D[m][n] = qn[m] + pn[n] - 2*(r_q r_p + c_q c_p) = squared distance
        v8f D = __builtin_amdgcn_wmma_f32_16x16x4_f32(
            false, A, false, B, (short)0, Cm, false, false);
#pragma unroll
        for (int v = 0; v < 8; ++v) minv[v] = fminf(minv[v], D[v]);
    }

    // Transpose partial minima through LDS, reduce each row's 16 columns.
#pragma unroll
    for (int v = 0; v < 8; ++v) s_min[(v + 8 * hi) * 16 + nl] = minv[v];
    __syncthreads();

    if (lane < 16) {
        float rm = FINF;
#pragma unroll
        for (int k = 0; k < 16; ++k) rm = fminf(rm, s_min[lane * 16 + k]);
        int gq = tile * 16 + lane;
        // fmaxf guards tiny negative d^2 from rounding when q == p
        s_qn[lane] = (gq < cq) ? sqrtf(fmaxf(rm, 0.0f)) : 0.0f;
    }
    __syncthreads();
    if (lane == 0) {
        float sum = 0.0f;
        for (int k = 0; k < 16; ++k) sum += s_qn[k];
        partials[y * QT_ + tile] = sum;     // fixed slot -> deterministic
    }
}

__global__ void hd_finalize_kernel(const float* __restrict__ partials,
                                   const int* __restrict__ counts,
                                   float* __restrict__ out) {
    __shared__ float tot[12];
    int t = threadIdx.x;
    if (t < 12) {
        float s = 0.0f;
        for (int i = 0; i < QT_; ++i) s += partials[t * QT_ + i];  // fixed order
        tot[t] = s;
    }
    __syncthreads();
    if (t == 0) {
        float fwd[NPAIR], rev[NPAIR];
        for (int p = 0; p < NPAIR; ++p) {
            int ca = counts[p]         > 0 ? counts[p]         : 1;
            int cb = counts[NPAIR + p] > 0 ? counts[NPAIR + p] : 1;
            fwd[p] = tot[p * 2 + 0] / (float)ca;
            rev[p] = tot[p * 2 + 1] / (float)cb;
        }
        float mhd[C_ + 2], fhd[C_ + 2], rhd[C_ + 2];
        for (int j = 0; j < C_ + 2; ++j) { mhd[j] = 0.f; fhd[j] = 0.f; rhd[j] = 0.f; }
        for (int b = 0; b < N_; ++b)
            for (int j = 1; j < C_; ++j) {
                int p = b * (C_ - 1) + (j - 1);
                fhd[j] += fwd[p];
                rhd[j] += rev[p];
                mhd[j] += fmaxf(fwd[p], rev[p]);
            }
        float* arrs[3] = { mhd, fhd, rhd };      // return order: mhd, fhd, rhd
        for (int a = 0; a < 3; ++a) {
            float* x = arrs[a];
            float s4 = 0.0f;
            for (int j = 0; j < C_; ++j) { x[j] /= (float)N_; s4 += x[j]; }
            x[C_]     = s4 / (float)C_;
            x[C_ + 1] = (x[1] + x[2] + x[3]) / 3.0f;
            for (int j = 0; j < C_ + 2; ++j) out[a * (C_ + 2) + j] = x[j];
        }
    }
}

extern "C" void kernel_launch(void* const* d_in, const int* in_sizes, int n_in,
                              void* d_out, int out_size, void* d_ws, size_t ws_size,
                              hipStream_t stream) {
    (void)in_sizes; (void)n_in; (void)out_size; (void)ws_size;
    const float* pred   = (const float*)d_in[0];
    const int*   labels = (const int*)d_in[1];

    int*    counts   = (int*)d_ws;
    float*  partials = (float*)((char*)d_ws + 256);
    float2* lists    = (float2*)((char*)d_ws + 256 + 12 * QT_ * (int)sizeof(float)); // @37888

    hd_init_kernel<<<(12 * HW_ + 255) / 256, 256, 0, stream>>>(lists, counts);
    hd_compact_kernel<<<(N_ * HW_ + 255) / 256, 256, 0, stream>>>(pred, labels, lists, counts);
    dim3 grid(QT_, 12);
    hd_min_kernel<<<grid, 32, 0, stream>>>(lists, counts, partials);
    hd_finalize_kernel<<<1, 32, 0, stream>>>(partials, counts, (float*)d_out);
}